// GinNet_15358803050743
// MI455X (gfx1250) — compile-verified
//
#include <hip/hip_runtime.h>
#include <hip/hip_bf16.h>

#define HID 64
#define TS  68   // LDS tile row stride in floats (64 + 4 pad; keeps rows 16B-aligned for async b128)

typedef float v2f __attribute__((ext_vector_type(2)));
typedef float v8f __attribute__((ext_vector_type(8)));

// ---------------------------------------------------------------- utilities

// agg[i] = (1+eps) * h[i]  (float4-vectorized), and zero the stats buffer.
// After this, edge atomics accumulate on top, so agg becomes z = (1+eps)h + sum_neigh.
__global__ void scale64_kernel(const float* __restrict__ h, float* __restrict__ agg,
                               const float* __restrict__ epsArr, int epsIdx,
                               float* __restrict__ stats, int total4) {
    int i = blockIdx.x * blockDim.x + threadIdx.x;
    if (i < 2 * HID) stats[i] = 0.0f;
    if (i >= total4) return;
    const float ep1 = 1.0f + epsArr[epsIdx];
    float4 v = ((const float4*)h)[i];
    float4 o;
    o.x = ep1 * v.x; o.y = ep1 * v.y; o.z = ep1 * v.z; o.w = ep1 * v.w;
    ((float4*)agg)[i] = o;
}

// layer-0 variant: scalar features. agg[n] = (1+eps0)*x[n]; zero stats.
__global__ void scale1_kernel(const float* __restrict__ x, float* __restrict__ agg,
                              const float* __restrict__ epsArr, float* __restrict__ stats,
                              int n) {
    int i = blockIdx.x * blockDim.x + threadIdx.x;
    if (i < 2 * HID) stats[i] = 0.0f;
    if (i < n) agg[i] = (1.0f + epsArr[0]) * x[i];
}

// scalar aggregation for layer 0 (features = 1)
__global__ void agg1_kernel(const float* __restrict__ x, const int* __restrict__ src,
                            const int* __restrict__ dst, float* __restrict__ agg, int nE) {
    int e = blockIdx.x * blockDim.x + threadIdx.x;
    if (e < nE) atomicAdd(&agg[dst[e]], x[src[e]]);
}

// 64-channel aggregation: one thread per (edge, 4 channels)
__global__ void agg64_kernel(const float* __restrict__ h, const int* __restrict__ src,
                             const int* __restrict__ dst, float* __restrict__ agg, int nE) {
    int t = blockIdx.x * blockDim.x + threadIdx.x;
    int e = t >> 4;
    if (e >= nE) return;
    int g = (t & 15) * 4;
    const float4 v = *(const float4*)&h[(size_t)src[e] * HID + g];
    float* p = &agg[(size_t)dst[e] * HID + g];
    atomicAdd(p + 0, v.x); atomicAdd(p + 1, v.y);
    atomicAdd(p + 2, v.z); atomicAdd(p + 3, v.w);
}

// ------------------------------------------------ wave-level 16x64 @ 64x64 WMMA

// tz: 16x64 fp32 A-tile in LDS (stride TS). sW: 64x64 row-major weights in LDS.
// Result: acc[t] = relu( tile @ W + bias ) for the four 16x16 N-tiles.
__device__ __forceinline__ void wave_mm64(const float* tz, const float* sW,
                                          const float* __restrict__ bias,
                                          int lane, v8f acc[4]) {
    const int half = lane >> 4;     // 0: lanes 0-15, 1: lanes 16-31
    const int l16  = lane & 15;
    v8f zero = {};
    #pragma unroll
    for (int t = 0; t < 4; ++t) acc[t] = zero;

    #pragma unroll
    for (int s = 0; s < 16; ++s) {          // K = 64 in steps of 4
        const int k = 4 * s + 2 * half;     // A layout: lanes 0-15 -> K=k0,k0+1; 16-31 -> k0+2,k0+3
        v2f a;
        a.x = tz[l16 * TS + k];
        a.y = tz[l16 * TS + k + 1];
        #pragma unroll
        for (int t = 0; t < 4; ++t) {
            const int col = t * 16 + l16;
            v2f b;
            b.x = sW[k * HID + col];
            b.y = sW[(k + 1) * HID + col];
            acc[t] = __builtin_amdgcn_wmma_f32_16x16x4_f32(
                false, a, false, b, (short)0, acc[t], false, false);
        }
    }
    // bias + relu (D layout: vgpr r -> M=r (lanes 0-15) / M=8+r (lanes 16-31), N=l16)
    #pragma unroll
    for (int t = 0; t < 4; ++t) {
        const float bv = bias[t * 16 + l16];
        #pragma unroll
        for (int r = 0; r < 8; ++r) {
            float v = acc[t][r] + bv;
            acc[t][r] = v > 0.0f ? v : 0.0f;
        }
    }
}

__device__ __forceinline__ void wave_store_tile_lds(float* tz, const v8f acc[4], int lane) {
    const int half = lane >> 4, l16 = lane & 15;
    #pragma unroll
    for (int t = 0; t < 4; ++t)
        #pragma unroll
        for (int r = 0; r < 8; ++r) {
            const int row = half ? (8 + r) : r;
            tz[row * TS + t * 16 + l16] = acc[t][r];
        }
}

// store result tile + accumulate per-channel sum/sumsq into block-shared stats
__device__ __forceinline__ void wave_emit(const v8f acc[4], float* __restrict__ hOut,
                                          float* sStats, int node0, int lane) {
    const int half = lane >> 4, l16 = lane & 15;
    #pragma unroll
    for (int t = 0; t < 4; ++t) {
        const int col = t * 16 + l16;
        float s = 0.0f, s2 = 0.0f;
        #pragma unroll
        for (int r = 0; r < 8; ++r) {
            const int row = half ? (8 + r) : r;
            const float v = acc[t][r];
            hOut[(size_t)(node0 + row) * HID + col] = v;
            s += v; s2 += v * v;
        }
        atomicAdd(&sStats[col], s);          // ds_add_f32 (LDS atomic)
        atomicAdd(&sStats[HID + col], s2);
    }
}

// async-copy one 16x64 f32 tile (row-contiguous float4 per lane) from global into LDS
__device__ __forceinline__ void async_stage_tile(float* tz, const float* __restrict__ gsrc,
                                                 int node0, int lane) {
    const int half = lane >> 4, l16 = lane & 15;
    const unsigned ldsBase = (unsigned)(uintptr_t)tz;   // LDS aperture: low 32 bits = LDS addr
    #pragma unroll
    for (int i = 0; i < 8; ++i) {
        const int row = i * 2 + half;
        const float* g = &gsrc[(size_t)(node0 + row) * HID + l16 * 4];
        const unsigned l = ldsBase + (unsigned)(row * TS + l16 * 4) * 4u;
        asm volatile("global_load_async_to_lds_b128 %0, %1, off"
                     :: "v"(l), "v"(g) : "memory");
    }
    asm volatile("s_wait_asynccnt 0x0" ::: "memory");
}

// ---------------------------------------------------------- GIN layer kernels

// Layers 1-3: z already in agg; relu(z@W1+b1)@W2+b2 ; relu ; stats
__global__ void __launch_bounds__(128)
gin_mlp_kernel(const float* __restrict__ agg,
               const float* __restrict__ w1, const float* __restrict__ b1,
               const float* __restrict__ w2, const float* __restrict__ b2,
               float* __restrict__ hOut, float* __restrict__ stats, int nTiles) {
    __shared__ float sW1[HID * HID];
    __shared__ float sW2[HID * HID];
    __shared__ float sT[4][16 * TS];
    __shared__ float sStats[2 * HID];

    const int tid = threadIdx.x;
    sStats[tid] = 0.0f;                                  // blockDim == 128 == 2*HID
    for (int i = tid; i < HID * HID; i += 128) { sW1[i] = w1[i]; sW2[i] = w2[i]; }
    __syncthreads();

    const int wave = tid >> 5, lane = tid & 31;
    const int tileIdx = blockIdx.x * 4 + wave;
    const bool active = tileIdx < nTiles;
    const int tile = active ? tileIdx : (nTiles - 1);    // keep EXEC all-ones for WMMA
    const int node0 = tile * 16;
    float* tz = sT[wave];

    async_stage_tile(tz, agg, node0, lane);              // z tile -> LDS (ASYNCcnt path)

    v8f acc[4];
    wave_mm64(tz, sW1, b1, lane, acc);                   // hidden = relu(z@W1+b1)
    wave_store_tile_lds(tz, acc, lane);                  // hidden -> LDS (same-wave, in-order)
    wave_mm64(tz, sW2, b2, lane, acc);                   // out = relu(hidden@W2+b2)

    if (active) wave_emit(acc, hOut, sStats, node0, lane);
    __syncthreads();
    atomicAdd(&stats[tid], sStats[tid]);                 // one global atomic per channel/block
}

// Layer 0: z (scalar) already in agg; outer product in VALU, second matmul uses WMMA
__global__ void __launch_bounds__(128)
gin0_kernel(const float* __restrict__ aggz,
            const float* __restrict__ w1a, const float* __restrict__ b1a,
            const float* __restrict__ w2a, const float* __restrict__ b2a,
            float* __restrict__ hOut, float* __restrict__ stats, int nTiles) {
    __shared__ float sW2[HID * HID];
    __shared__ float sT[4][16 * TS];
    __shared__ float sStats[2 * HID];

    const int tid = threadIdx.x;
    sStats[tid] = 0.0f;
    for (int i = tid; i < HID * HID; i += 128) sW2[i] = w2a[i];
    __syncthreads();

    const int wave = tid >> 5, lane = tid & 31;
    const int half = lane >> 4, l16 = lane & 15;
    const int tileIdx = blockIdx.x * 4 + wave;
    const bool active = tileIdx < nTiles;
    const int tile = active ? tileIdx : (nTiles - 1);
    const int node0 = tile * 16;
    float* tz = sT[wave];

    #pragma unroll
    for (int i = 0; i < 8; ++i) {
        const int row = i * 2 + half;
        const float z = aggz[node0 + row];
        const int c = l16 * 4;
        const float4 w = *(const float4*)&w1a[c];
        const float4 b = *(const float4*)&b1a[c];
        float* d = &tz[row * TS + c];
        float v0 = z * w.x + b.x; d[0] = v0 > 0.0f ? v0 : 0.0f;
        float v1 = z * w.y + b.y; d[1] = v1 > 0.0f ? v1 : 0.0f;
        float v2 = z * w.z + b.z; d[2] = v2 > 0.0f ? v2 : 0.0f;
        float v3 = z * w.w + b.w; d[3] = v3 > 0.0f ? v3 : 0.0f;
    }

    v8f acc[4];
    wave_mm64(tz, sW2, b2a, lane, acc);
    if (active) wave_emit(acc, hOut, sStats, node0, lane);
    __syncthreads();
    atomicAdd(&stats[tid], sStats[tid]);
}

// BatchNorm apply (stats hold per-channel sum / sumsq post-relu), float4-vectorized
__global__ void bn_kernel(float* __restrict__ h, const float* __restrict__ stats,
                          const float* __restrict__ gamma, const float* __restrict__ beta,
                          int nNodes, int total4) {
    const int i = blockIdx.x * blockDim.x + threadIdx.x;
    if (i >= total4) return;
    const int c = (i * 4) & (HID - 1);
    const float invN = 1.0f / (float)nNodes;
    float4 v = ((const float4*)h)[i];
    float* vp = (float*)&v;
    #pragma unroll
    for (int j = 0; j < 4; ++j) {
        const float mu  = stats[c + j] * invN;
        const float var = stats[HID + c + j] * invN - mu * mu;
        vp[j] = (vp[j] - mu) * rsqrtf(var + 1e-5f) * gamma[c + j] + beta[c + j];
    }
    ((float4*)h)[i] = v;
}

__global__ void fc_kernel(const float* __restrict__ h, const float* __restrict__ w,
                          const float* __restrict__ b, float* __restrict__ out, int nNodes) {
    const int n = blockIdx.x * blockDim.x + threadIdx.x;
    if (n >= nNodes) return;
    float acc = b[0];
    #pragma unroll
    for (int c = 0; c < HID; ++c) acc += h[(size_t)n * HID + c] * w[c];
    out[n] = acc;
}

// ------------------------------------------------------------------- launcher

extern "C" void kernel_launch(void* const* d_in, const int* in_sizes, int n_in,
                              void* d_out, int out_size, void* d_ws, size_t ws_size,
                              hipStream_t stream) {
    const float* x      = (const float*)d_in[0];
    const int*   ei     = (const int*)  d_in[1];
    const float* w1a    = (const float*)d_in[2];
    const float* b1a    = (const float*)d_in[3];
    const float* w2a    = (const float*)d_in[4];
    const float* b2a    = (const float*)d_in[5];
    const float* w1s    = (const float*)d_in[6];
    const float* b1s    = (const float*)d_in[7];
    const float* w2s    = (const float*)d_in[8];
    const float* b2s    = (const float*)d_in[9];
    const float* eps    = (const float*)d_in[10];
    const float* gammas = (const float*)d_in[11];
    const float* betas  = (const float*)d_in[12];
    const float* fcw    = (const float*)d_in[13];
    const float* fcb    = (const float*)d_in[14];

    const int N = in_sizes[0];          // 50000 (divisible by 16)
    const int E = in_sizes[1] / 2;      // 1,600,000
    const int* src = ei;
    const int* dst = ei + E;

    float* hA    = (float*)d_ws;
    float* hB    = hA + (size_t)N * HID;
    float* agg   = hB + (size_t)N * HID;
    float* stats = agg + (size_t)N * HID;   // 128 floats: sum[64], sumsq[64]

    const int nTiles    = (N + 15) / 16;
    const int mlpBlocks = (nTiles + 3) / 4;
    const int total4    = N * HID / 4;
    const int v4Blocks  = (total4 + 255) / 256;

    // ---- layer 0 (1 -> 64): z(scalar) = (1+eps0)x + neighbor-sum, built in agg ----
    scale1_kernel<<<(N + 255) / 256, 256, 0, stream>>>(x, agg, eps, stats, N);
    agg1_kernel<<<(E + 255) / 256, 256, 0, stream>>>(x, src, dst, agg, E);
    gin0_kernel<<<mlpBlocks, 128, 0, stream>>>(agg, w1a, b1a, w2a, b2a,
                                               hA, stats, nTiles);
    bn_kernel<<<v4Blocks, 256, 0, stream>>>(hA, stats, gammas, betas, N, total4);

    // ---- layers 1-3 (64 -> 64): agg = (1+eps)h, then += neighbor-sum => z ----
    const float* hin = hA;
    float* hout = hB;
    for (int i = 0; i < 3; ++i) {
        scale64_kernel<<<v4Blocks, 256, 0, stream>>>(hin, agg, eps, i + 1, stats, total4);
        agg64_kernel<<<(E * 16 + 255) / 256, 256, 0, stream>>>(hin, src, dst, agg, E);
        gin_mlp_kernel<<<mlpBlocks, 128, 0, stream>>>(agg,
            w1s + (size_t)i * HID * HID, b1s + i * HID,
            w2s + (size_t)i * HID * HID, b2s + i * HID,
            hout, stats, nTiles);
        bn_kernel<<<v4Blocks, 256, 0, stream>>>(hout, stats,
            gammas + (size_t)(i + 1) * HID, betas + (size_t)(i + 1) * HID, N, total4);
        float* tmp = (float*)hin; hin = hout; hout = tmp;
    }

    // ---- readout ----
    fc_kernel<<<(N + 255) / 256, 256, 0, stream>>>(hin, fcw, fcb, (float*)d_out, N);
    (void)n_in; (void)out_size; (void)ws_size;
}